// TEXTGCN_44152263803038
// MI455X (gfx1250) — compile-verified
//
#include <hip/hip_runtime.h>
#include <hip/hip_bf16.h>

typedef float v2f __attribute__((ext_vector_type(2)));
typedef float v8f __attribute__((ext_vector_type(8)));

#define N_NODES 100000
#define N_EDGES 1600000
#define DIM_IN  256
#define HD      64
#define NCLS    20
#define NDOCS   10000

__device__ __forceinline__ void atomic_add_f32(float* p, float v) {
    // relaxed, agent scope -> global_atomic_add_f32 (no CAS loop)
    __hip_atomic_fetch_add(p, v, __ATOMIC_RELAXED, __HIP_MEMORY_SCOPE_AGENT);
}

// ---------------- degree / normalization ----------------

__global__ void TEXTGCN_deg_init(float* __restrict__ deg, int n) {
    int i = blockIdx.x * blockDim.x + threadIdx.x;
    if (i < n) deg[i] = 1.0f;                      // self-loop weight
}

__global__ void TEXTGCN_deg_acc(const int* __restrict__ dst,
                                const float* __restrict__ w,
                                float* __restrict__ deg, int e) {
    int i = blockIdx.x * blockDim.x + threadIdx.x;
    if (i < e) atomic_add_f32(&deg[dst[i]], w[i]);
}

__global__ void TEXTGCN_dinv(float* __restrict__ deg, int n) {
    int i = blockIdx.x * blockDim.x + threadIdx.x;
    if (i < n) {
        float d = deg[i];
        deg[i] = (d > 0.0f) ? __frsqrt_rn(d) : 0.0f;   // in-place: deg -> dinv
    }
}

__global__ void TEXTGCN_norm(const int* __restrict__ src, const int* __restrict__ dst,
                             const float* __restrict__ w, const float* __restrict__ dinv,
                             float* __restrict__ norm, int e) {
    int i = blockIdx.x * blockDim.x + threadIdx.x;
    if (i < e) norm[i] = dinv[src[i]] * w[i] * dinv[dst[i]];
}

// ---------------- WMMA f32 GEMM: C[M x Nout] = A[M x K] @ B[K x Nout] ----------------
// one wave (32 threads) per 16x16 output tile, V_WMMA_F32_16X16X4_F32, K stepped by 4.

__global__ void TEXTGCN_gemm_wmma(const float* __restrict__ A,
                                  const float* __restrict__ B,
                                  float* __restrict__ C,
                                  int K, int Nout) {
    const int lane = threadIdx.x;                 // blockDim.x == 32
    const int mt = blockIdx.x, nt = blockIdx.y;
    const int m  = mt * 16 + (lane & 15);
    const int n  = nt * 16 + (lane & 15);
    const int kq = (lane >> 4) * 2;               // lanes 0-15: K 0,1 ; lanes 16-31: K 2,3

    const float* __restrict__ arow = A + (size_t)m * K;
    v8f acc = {};
    for (int k = 0; k < K; k += 4) {
        v2f a, b;
        a.x = arow[k + kq];
        a.y = arow[k + kq + 1];
        b.x = B[(size_t)(k + kq)     * Nout + n];
        b.y = B[(size_t)(k + kq + 1) * Nout + n];
        acc = __builtin_amdgcn_wmma_f32_16x16x4_f32(
                  false, a, false, b, (short)0, acc, false, false);
    }
    // C/D layout: VGPR r holds M = r (lanes 0-15) / r+8 (lanes 16-31), N = lane&15
    const int mbase = mt * 16 + (lane >> 4) * 8;
    const int nst   = nt * 16 + (lane & 15);
#pragma unroll
    for (int r = 0; r < 8; ++r)
        C[(size_t)(mbase + r) * Nout + nst] = acc[r];
}

// ---------------- graph propagate: acc = scatter(t * norm) ----------------

__global__ void TEXTGCN_self(const float* __restrict__ t, const float* __restrict__ dinv,
                             float* __restrict__ acc, int n) {
    int gid = blockIdx.x * blockDim.x + threadIdx.x;        // n * 16 chunks of float4
    if (gid >= n * (HD / 4)) return;
    int i = gid >> 4, c = gid & 15;
    float s = dinv[i] * dinv[i];                            // self-loop norm = 1/deg
    float4 v = ((const float4*)(t + (size_t)i * HD))[c];
    v.x *= s; v.y *= s; v.z *= s; v.w *= s;
    ((float4*)(acc + (size_t)i * HD))[c] = v;
}

__global__ void TEXTGCN_scatter(const int* __restrict__ src, const int* __restrict__ dst,
                                const float* __restrict__ norm,
                                const float* __restrict__ t,
                                float* __restrict__ acc, int e) {
    int gid = blockIdx.x * blockDim.x + threadIdx.x;        // e * 16 chunks of float4
    if (gid >= e * (HD / 4)) return;
    int ei = gid >> 4, c = gid & 15;
    int s = src[ei], d = dst[ei];
    float w = norm[ei];
    float4 v = ((const float4*)(t + (size_t)s * HD))[c];
    float* o = acc + (size_t)d * HD + c * 4;
    atomic_add_f32(o + 0, v.x * w);
    atomic_add_f32(o + 1, v.y * w);
    atomic_add_f32(o + 2, v.z * w);
    atomic_add_f32(o + 3, v.w * w);
}

__global__ void TEXTGCN_bias_relu(float* __restrict__ acc, const float* __restrict__ b, int n) {
    int gid = blockIdx.x * blockDim.x + threadIdx.x;        // n * HD
    if (gid >= n * HD) return;
    float v = acc[gid] + b[gid & (HD - 1)];
    acc[gid] = v > 0.0f ? v : 0.0f;
}

// ---------------- doc pooling: dbuf = relu(h3[doc_idx] @ Wp + bp), WMMA fused gather ----------------

__global__ void TEXTGCN_doc_gemm_wmma(const float* __restrict__ h,
                                      const int* __restrict__ doc_idx,
                                      const float* __restrict__ Wp,
                                      const float* __restrict__ bp,
                                      float* __restrict__ dbuf) {
    const int lane = threadIdx.x;
    const int mt = blockIdx.x, nt = blockIdx.y;
    const int m  = mt * 16 + (lane & 15);
    const int n  = nt * 16 + (lane & 15);
    const int kq = (lane >> 4) * 2;

    const float* __restrict__ arow = h + (size_t)doc_idx[m] * HD;
    v8f acc = {};
    for (int k = 0; k < HD; k += 4) {
        v2f a, b;
        a.x = arow[k + kq];
        a.y = arow[k + kq + 1];
        b.x = Wp[(size_t)(k + kq)     * HD + n];
        b.y = Wp[(size_t)(k + kq + 1) * HD + n];
        acc = __builtin_amdgcn_wmma_f32_16x16x4_f32(
                  false, a, false, b, (short)0, acc, false, false);
    }
    const int mbase = mt * 16 + (lane >> 4) * 8;
    const int nst   = nt * 16 + (lane & 15);
    const float bias = bp[nst];
#pragma unroll
    for (int r = 0; r < 8; ++r) {
        float v = acc[r] + bias;
        dbuf[(size_t)(mbase + r) * HD + nst] = v > 0.0f ? v : 0.0f;
    }
}

// ---------------- classifier: out = dbuf @ Wc + bc  (10000 x 20, tiny) ----------------

__global__ void TEXTGCN_classifier(const float* __restrict__ dbuf,
                                   const float* __restrict__ Wc,
                                   const float* __restrict__ bc,
                                   float* __restrict__ out) {
    int gid = blockIdx.x * blockDim.x + threadIdx.x;
    if (gid >= NDOCS * NCLS) return;
    int d = gid / NCLS, c = gid % NCLS;
    const float* __restrict__ row = dbuf + (size_t)d * HD;
    float s = bc[c];
#pragma unroll 8
    for (int k = 0; k < HD; ++k) s += row[k] * Wc[(size_t)k * NCLS + c];
    out[gid] = s;
}

// ---------------- host launcher ----------------

extern "C" void kernel_launch(void* const* d_in, const int* in_sizes, int n_in,
                              void* d_out, int out_size, void* d_ws, size_t ws_size,
                              hipStream_t stream) {
    const float* x   = (const float*)d_in[0];
    const float* ew  = (const float*)d_in[1];
    const float* W1  = (const float*)d_in[2];
    const float* b1  = (const float*)d_in[3];
    const float* W2  = (const float*)d_in[4];
    const float* b2  = (const float*)d_in[5];
    const float* W3  = (const float*)d_in[6];
    const float* b3  = (const float*)d_in[7];
    const float* Wp  = (const float*)d_in[8];
    const float* bp  = (const float*)d_in[9];
    const float* Wc  = (const float*)d_in[10];
    const float* bc  = (const float*)d_in[11];
    const int*   ei  = (const int*)d_in[12];     // [2, E]: src row then dst row
    const int*   doc = (const int*)d_in[13];
    float*       out = (float*)d_out;

    const int* src = ei;
    const int* dst = ei + N_EDGES;

    // workspace layout (floats)
    float* ws   = (float*)d_ws;
    float* dinv = ws;                            // N  (deg -> dinv in place)
    float* nrm  = dinv + N_NODES;                // E
    float* bufT = nrm  + N_EDGES;                // N*64  gemm output
    float* bufH = bufT + (size_t)N_NODES * HD;   // N*64  acc / hidden
    float* dbuf = bufH + (size_t)N_NODES * HD;   // NDOC*64

    const int B = 256;
    dim3 blk(B);
    dim3 gN((N_NODES + B - 1) / B);
    dim3 gE((N_EDGES + B - 1) / B);
    dim3 gN16((N_NODES * (HD / 4) + B - 1) / B);
    dim3 gE16((N_EDGES * (HD / 4) + B - 1) / B);
    dim3 gNF((N_NODES * HD + B - 1) / B);

    // normalization
    TEXTGCN_deg_init<<<gN, blk, 0, stream>>>(dinv, N_NODES);
    TEXTGCN_deg_acc <<<gE, blk, 0, stream>>>(dst, ew, dinv, N_EDGES);
    TEXTGCN_dinv    <<<gN, blk, 0, stream>>>(dinv, N_NODES);
    TEXTGCN_norm    <<<gE, blk, 0, stream>>>(src, dst, ew, dinv, nrm, N_EDGES);

    dim3 wave(32);
    dim3 gGemm1(N_NODES / 16, HD / 16);          // 6250 x 4
    dim3 gGemmH(N_NODES / 16, HD / 16);
    dim3 gDoc(NDOCS / 16, HD / 16);              // 625 x 4

    // conv1: t = x @ W1 ; h = relu(prop(t) + b1)
    TEXTGCN_gemm_wmma<<<gGemm1, wave, 0, stream>>>(x, W1, bufT, DIM_IN, HD);
    TEXTGCN_self     <<<gN16, blk, 0, stream>>>(bufT, dinv, bufH, N_NODES);
    TEXTGCN_scatter  <<<gE16, blk, 0, stream>>>(src, dst, nrm, bufT, bufH, N_EDGES);
    TEXTGCN_bias_relu<<<gNF,  blk, 0, stream>>>(bufH, b1, N_NODES);

    // conv2
    TEXTGCN_gemm_wmma<<<gGemmH, wave, 0, stream>>>(bufH, W2, bufT, HD, HD);
    TEXTGCN_self     <<<gN16, blk, 0, stream>>>(bufT, dinv, bufH, N_NODES);
    TEXTGCN_scatter  <<<gE16, blk, 0, stream>>>(src, dst, nrm, bufT, bufH, N_EDGES);
    TEXTGCN_bias_relu<<<gNF,  blk, 0, stream>>>(bufH, b2, N_NODES);

    // conv3
    TEXTGCN_gemm_wmma<<<gGemmH, wave, 0, stream>>>(bufH, W3, bufT, HD, HD);
    TEXTGCN_self     <<<gN16, blk, 0, stream>>>(bufT, dinv, bufH, N_NODES);
    TEXTGCN_scatter  <<<gE16, blk, 0, stream>>>(src, dst, nrm, bufT, bufH, N_EDGES);
    TEXTGCN_bias_relu<<<gNF,  blk, 0, stream>>>(bufH, b3, N_NODES);

    // doc head
    TEXTGCN_doc_gemm_wmma<<<gDoc, wave, 0, stream>>>(bufH, doc, Wp, bp, dbuf);
    dim3 gCls((NDOCS * NCLS + B - 1) / B);
    TEXTGCN_classifier<<<gCls, blk, 0, stream>>>(dbuf, Wc, bc, out);
}